// FullGraphModel_292057776280
// MI455X (gfx1250) — compile-verified
//
#include <hip/hip_runtime.h>
#include <hip/hip_bf16.h>

typedef __attribute__((ext_vector_type(2))) float v2f;
typedef __attribute__((ext_vector_type(8))) float v8f;

#define BB 16           // batch == WMMA M dimension
#define FCD 512         // D
#define FCC 10          // C
#define RB 1024         // reduction partial blocks
#define NUM_PASSES 4

// ---- w[e] = edge_weight[e] * tanh(mult[e]) ------------------------------
__global__ void prep_w_kernel(const float* __restrict__ ew,
                              const float* __restrict__ mult,
                              float* __restrict__ w, int E) {
    int stride = gridDim.x * blockDim.x;
    for (int e = blockIdx.x * blockDim.x + threadIdx.x; e < E; e += stride)
        w[e] = ew[e] * tanhf(mult[e]);
}

// ---- xb_t[n*16 + b] = x[b*N + n] ----------------------------------------
__global__ void transpose_x_kernel(const float* __restrict__ x,
                                   float* __restrict__ xbt, int N) {
    int total = N * BB;
    int stride = gridDim.x * blockDim.x;
    for (int i = blockIdx.x * blockDim.x + threadIdx.x; i < total; i += stride) {
        int n = i >> 4, b = i & 15;
        xbt[i] = x[b * N + n];
    }
}

// ---- edge gather-scatter: 2 edges per wave32 (16 lanes each) ------------
__global__ void edge_scatter_kernel(const int* __restrict__ ei,
                                    const float* __restrict__ w,
                                    const float* __restrict__ xbt,
                                    float* __restrict__ aggr, int E) {
    int tid = blockIdx.x * blockDim.x + threadIdx.x;
    int b   = tid & 15;          // batch lane within half-wave
    int hw  = tid >> 4;          // half-wave id == edge slot
    int nhw = (gridDim.x * blockDim.x) >> 4;
    for (int e = hw; e < E; e += nhw) {
        int   src = ei[e];
        int   dst = ei[E + e];
        float v   = xbt[src * BB + b] * w[e];
        unsafeAtomicAdd(&aggr[dst * BB + b], v);   // -> global_atomic_add_f32
    }
}

// ---- global min/max, two-stage ------------------------------------------
__global__ void minmax_partial_kernel(const float* __restrict__ a,
                                      float* __restrict__ pmin,
                                      float* __restrict__ pmax, int n) {
    __shared__ float smin[256];
    __shared__ float smax[256];
    float lmin = 3.402823466e38f, lmax = -3.402823466e38f;
    int stride = gridDim.x * blockDim.x;
    for (int i = blockIdx.x * blockDim.x + threadIdx.x; i < n; i += stride) {
        float v = a[i];
        lmin = fminf(lmin, v);
        lmax = fmaxf(lmax, v);
    }
    int t = threadIdx.x;
    smin[t] = lmin; smax[t] = lmax;
    __syncthreads();
    for (int s = 128; s > 0; s >>= 1) {
        if (t < s) {
            smin[t] = fminf(smin[t], smin[t + s]);
            smax[t] = fmaxf(smax[t], smax[t + s]);
        }
        __syncthreads();
    }
    if (t == 0) { pmin[blockIdx.x] = smin[0]; pmax[blockIdx.x] = smax[0]; }
}

__global__ void minmax_final_kernel(const float* __restrict__ pmin,
                                    const float* __restrict__ pmax,
                                    float* __restrict__ mm, int nb) {
    __shared__ float smin[256];
    __shared__ float smax[256];
    float lmin = 3.402823466e38f, lmax = -3.402823466e38f;
    int t = threadIdx.x;
    for (int i = t; i < nb; i += 256) {
        lmin = fminf(lmin, pmin[i]);
        lmax = fmaxf(lmax, pmax[i]);
    }
    smin[t] = lmin; smax[t] = lmax;
    __syncthreads();
    for (int s = 128; s > 0; s >>= 1) {
        if (t < s) {
            smin[t] = fminf(smin[t], smin[t + s]);
            smax[t] = fmaxf(smax[t], smax[t + s]);
        }
        __syncthreads();
    }
    if (t == 0) { mm[0] = smin[0]; mm[1] = smax[0]; }
}

// ---- sigmoid update; re-zeroes aggr for next pass -----------------------
__global__ void update_kernel(float* __restrict__ aggr,
                              const float* __restrict__ thr,
                              float* __restrict__ xbt,
                              const float* __restrict__ mm, int total) {
    float amin = mm[0];
    float rs   = 1.0f / (mm[1] - amin);
    int stride = gridDim.x * blockDim.x;
    for (int i = blockIdx.x * blockDim.x + threadIdx.x; i < total; i += stride) {
        float z = (aggr[i] - amin) * rs - fabsf(thr[i >> 4]);
        xbt[i]  = 1.0f / (1.0f + __expf(-z));
        aggr[i] = 0.0f;
    }
}

// ---- final FC with f32 WMMA: out(16x10) = xd(16x512) @ fc_w^T + b -------
// LDS layout is k-pair-major: element (k, m) lives at ((k>>1)*16 + m)*2 + (k&1)
// so each lane's two K values are adjacent -> one ds_load_b64 per operand.
// A layout (16x4 f32, ISA 7.12.2): lane = (K/2)*16 + M; v0 = K&~1, v1 = K|1
// B layout (4x16 f32, mirrored):   lane = (K/2)*16 + N  (cols 10..15 zero-padded)
// C/D layout: VGPR v, lanes 0-15 -> row v, lanes 16-31 -> row v+8, col = lane&15
__global__ void fc_wmma_kernel(const float* __restrict__ xbt,
                               const int* __restrict__ dec,
                               const float* __restrict__ fcw,
                               const float* __restrict__ fcb,
                               float* __restrict__ out) {
    __shared__ float xdT[FCD * BB];    // 32 KB, k-pair-major [p][m][2]
    __shared__ float fwT[FCD * BB];    // 32 KB, k-pair-major [p][n][2], zero-padded
    int t = threadIdx.x;
    if (t < FCD) {
        int k    = t;
        int base = ((k >> 1) * BB) * 2 + (k & 1);   // + 2*col
        int idx  = dec[k];
        #pragma unroll
        for (int b = 0; b < BB; ++b)
            xdT[base + 2 * b] = xbt[idx * BB + b];
        #pragma unroll
        for (int n = 0; n < BB; ++n)
            fwT[base + 2 * n] = (n < FCC) ? fcw[n * FCD + k] : 0.0f;
    }
    __syncthreads();
    if (t < 32) {
        int m  = t & 15;            // M for A, N for B, col for D
        int p0 = t >> 4;            // lanes 16-31 take the odd k-pair (K offs 2,3)
        const v2f* xa = (const v2f*)xdT;
        const v2f* xw = (const v2f*)fwT;
        v8f acc = {0.f, 0.f, 0.f, 0.f, 0.f, 0.f, 0.f, 0.f};
        #pragma unroll 16
        for (int k0 = 0; k0 < FCD; k0 += 4) {
            int p  = (k0 >> 1) + p0;
            v2f a  = xa[p * BB + m];
            v2f bm = xw[p * BB + m];
            acc = __builtin_amdgcn_wmma_f32_16x16x4_f32(
                false, a, false, bm, (short)0, acc, false, false);
        }
        if (m < FCC) {
            int   rowbase = p0 * 8;
            float bias    = fcb[m];
            #pragma unroll
            for (int v = 0; v < 8; ++v)
                out[(rowbase + v) * FCC + m] = acc[v] + bias;
        }
    }
}

extern "C" void kernel_launch(void* const* d_in, const int* in_sizes, int n_in,
                              void* d_out, int out_size, void* d_ws, size_t ws_size,
                              hipStream_t stream) {
    const float* x    = (const float*)d_in[0];
    const float* ew   = (const float*)d_in[1];
    const float* mult = (const float*)d_in[2];
    const float* thr  = (const float*)d_in[3];
    const float* fcw  = (const float*)d_in[4];
    const float* fcb  = (const float*)d_in[5];
    const int*   ei   = (const int*)d_in[6];
    const int*   dec  = (const int*)d_in[7];
    float* out = (float*)d_out;

    const int E     = in_sizes[1];
    const int N     = in_sizes[3];
    const int total = N * BB;

    float* wsf  = (float*)d_ws;
    float* w    = wsf;              // E floats
    float* xbt  = w + E;            // N*16 floats
    float* aggr = xbt + total;      // N*16 floats
    float* pmin = aggr + total;     // RB floats
    float* pmax = pmin + RB;        // RB floats
    float* mm   = pmax + RB;        // 2 floats

    hipMemsetAsync(aggr, 0, (size_t)total * sizeof(float), stream);
    prep_w_kernel<<<1024, 256, 0, stream>>>(ew, mult, w, E);
    transpose_x_kernel<<<1024, 256, 0, stream>>>(x, xbt, N);

    for (int p = 0; p < NUM_PASSES; ++p) {
        edge_scatter_kernel<<<2048, 256, 0, stream>>>(ei, w, xbt, aggr, E);
        minmax_partial_kernel<<<RB, 256, 0, stream>>>(aggr, pmin, pmax, total);
        minmax_final_kernel<<<1, 256, 0, stream>>>(pmin, pmax, mm, RB);
        update_kernel<<<1024, 256, 0, stream>>>(aggr, thr, xbt, mm, total);
    }

    fc_wmma_kernel<<<1, FCD, 0, stream>>>(xbt, dec, fcw, fcb, out);
}